// TwoBodyJastrowFactor_15728170238293
// MI455X (gfx1250) — compile-verified
//
#include <hip/hip_runtime.h>
#include <hip/hip_bf16.h>
#include <math.h>

typedef __attribute__((ext_vector_type(2))) float v2f;
typedef __attribute__((ext_vector_type(8))) float v8f;

#define NELEC 64
#define WPB 4   // waves (batch elements) per block

__global__ __launch_bounds__(32 * WPB)
void jastrow_grad_wmma(const float* __restrict__ pos,
                       const float* __restrict__ weight,
                       float* __restrict__ out, int nbatch)
{
    __shared__ float ldsX[WPB][NELEC * 3];

    const int lane = threadIdx.x & 31;
    const int wave = threadIdx.x >> 5;
    int b = blockIdx.x * WPB + wave;
    if (b >= nbatch) b = nbatch - 1;     // duplicate last element; deterministic

    const float w = weight[0];

    // Stage positions for this batch element into LDS (coalesced: 6 floats/lane).
    const float* px = pos + (size_t)b * (NELEC * 3);
    float* X = ldsX[wave];
#pragma unroll
    for (int k = 0; k < 6; ++k)
        X[lane + 32 * k] = px[lane + 32 * k];
    __syncthreads();

    const int  n  = lane & 15;
    const bool hi = lane >= 16;

    // WMMA f32 16x16x4 operand blocks (A layout == B layout for X / X^T):
    //   lanes 0-15 : v0 = X[row].x (K=0), v1 = X[row].y (K=1)
    //   lanes 16-31: v0 = X[row].z (K=2), v1 = 0        (K=3 pad)
    v2f blk[4];
#pragma unroll
    for (int bi = 0; bi < 4; ++bi) {
        int row = bi * 16 + n;
        float x0 = X[row * 3 + 0], x1 = X[row * 3 + 1], x2 = X[row * 3 + 2];
        blk[bi].x = hi ? x2 : x0;
        blk[bi].y = hi ? 0.0f : x1;
    }

    // Column data owned by this lane: j = bj*16 + n
    float cx[4][3], ncol[4];
#pragma unroll
    for (int bj = 0; bj < 4; ++bj) {
        int j = bj * 16 + n;
        cx[bj][0] = X[j * 3 + 0];
        cx[bj][1] = X[j * 3 + 1];
        cx[bj][2] = X[j * 3 + 2];
        ncol[bj] = cx[bj][0] * cx[bj][0] + cx[bj][1] * cx[bj][1] + cx[bj][2] * cx[bj][2];
    }

    float colT[4]    = {0.f, 0.f, 0.f, 0.f};
    float colTX[4][3] = {};
    float S = 0.0f;

#pragma unroll
    for (int bi = 0; bi < 4; ++bi) {
        // Row block for this lane-half: rows i0..i0+7 (uniform per half; LDS broadcast)
        int i0 = bi * 16 + (hi ? 8 : 0);
        float rx[8][3], nrow[8];
#pragma unroll
        for (int m = 0; m < 8; ++m) {
            rx[m][0] = X[(i0 + m) * 3 + 0];
            rx[m][1] = X[(i0 + m) * 3 + 1];
            rx[m][2] = X[(i0 + m) * 3 + 2];
            nrow[m] = rx[m][0] * rx[m][0] + rx[m][1] * rx[m][1] + rx[m][2] * rx[m][2];
        }
#pragma unroll
        for (int bj = 0; bj < 4; ++bj) {
            // G tile = X[bi-block] (16x4) * X[bj-block]^T (4x16), fp32 WMMA
            v8f c = {0.f, 0.f, 0.f, 0.f, 0.f, 0.f, 0.f, 0.f};
            c = __builtin_amdgcn_wmma_f32_16x16x4_f32(
                    /*neg_a=*/false, blk[bi], /*neg_b=*/false, blk[bj],
                    /*c_mod=*/(short)0, c, /*reuse_a=*/false, /*reuse_b=*/false);

            const float sw = ((bi < 2) == (bj < 2)) ? 0.25f : 0.5f; // spin blocks align to 32
            const int   j  = bj * 16 + n;
#pragma unroll
            for (int m = 0; m < 8; ++m) {
                int   i = i0 + m;
                float G = c[m];
                float r = nrow[m] + ncol[bj] - 2.0f * G;   // squared distance
                float denom = 1.0f + w * r;
                float rd = __builtin_amdgcn_rcpf(denom);
                // P exponent: sum sw*r/denom over i<j
                S += (i < j) ? (sw * r * rd) : 0.0f;
                // t[i,j] = sign(j-i)*sw / (r*denom^2)   (jast cancels against P/jast)
                float t = sw * __builtin_amdgcn_rcpf(r) * rd * rd;
                t = (i < j) ? t : -t;
                t = (i == j) ? 0.0f : t;
                colT[bj] += t;
                colTX[bj][0] += t * rx[m][0];
                colTX[bj][1] += t * rx[m][1];
                colTX[bj][2] += t * rx[m][2];
            }
        }
    }

    // Merge the two 8-row halves per column (antisymmetry: rowSum_j = -colSum_j)
#pragma unroll
    for (int bj = 0; bj < 4; ++bj) {
        colT[bj]     += __shfl_xor(colT[bj], 16, 32);
        colTX[bj][0] += __shfl_xor(colTX[bj][0], 16, 32);
        colTX[bj][1] += __shfl_xor(colTX[bj][1], 16, 32);
        colTX[bj][2] += __shfl_xor(colTX[bj][2], 16, 32);
    }
    // Full wave reduction of the pair-sum S
#pragma unroll
    for (int m = 16; m >= 1; m >>= 1)
        S += __shfl_xor(S, m, 32);

    const float P = __expf(S);   // single surviving transcendental per batch element

    if (lane < 16) {             // lanes 0-15 own all 16 columns per bj after merge
        float* o = out + (size_t)b * (NELEC * 3);
#pragma unroll
        for (int bj = 0; bj < 4; ++bj) {
            int j = bj * 16 + n;
#pragma unroll
            for (int d = 0; d < 3; ++d) {
                // g_j[d] = colTX - x_j*colT  (sign-folded row form)
                float g = colTX[bj][d] - cx[bj][d] * colT[bj];
                o[j * 3 + d] = 1.0f + P * g;
            }
        }
    }
}

extern "C" void kernel_launch(void* const* d_in, const int* in_sizes, int n_in,
                              void* d_out, int out_size, void* d_ws, size_t ws_size,
                              hipStream_t stream) {
    const float* pos    = (const float*)d_in[0];
    const float* weight = (const float*)d_in[1];
    float* out = (float*)d_out;
    int nbatch = in_sizes[0] / (NELEC * 3);   // 4096
    int blocks = (nbatch + WPB - 1) / WPB;
    jastrow_grad_wmma<<<blocks, 32 * WPB, 0, stream>>>(pos, weight, out, nbatch);
}